// Predict_Vectors_Down_26010321945024
// MI455X (gfx1250) — compile-verified
//
#include <hip/hip_runtime.h>
#include <stdint.h>

// out[gp*64 + oc] = sum_{k=0..15} W[gp*1024 + yx*256 + oc*4 + i] * x[gp*16 + yx*4 + i]
//   gp in [0, 65536), oc in [0,64), k = yx*4+i
// Pure streaming problem: 256MiB weights read once -> HBM bound (~12us at 23.3TB/s).
// fp32 throughout; K-reduction done on the matrix pipe via chained
// V_WMMA_F32_16X16X4_F32 with an all-ones B operand (exact f32 math).

typedef float v2f __attribute__((ext_vector_type(2)));
typedef float v4f __attribute__((ext_vector_type(4)));
typedef float v8f __attribute__((ext_vector_type(8)));

#define WAVES_PER_BLOCK 8
#define POS_PER_WAVE    4
#define POS_PER_BLOCK   (WAVES_PER_BLOCK * POS_PER_WAVE)   // 32 positions / block

__global__ __launch_bounds__(256)
void predict_vectors_wmma_kernel(const float* __restrict__ x,   // [GP,16]
                                 const float* __restrict__ w,   // [GP,4,64,4]
                                 float* __restrict__ out,       // [GP,64]
                                 int total_pos)
{
    __shared__ __align__(16) float xs[POS_PER_BLOCK * 16];      // 2KB

    const int wave = threadIdx.x >> 5;
    const int lane = threadIdx.x & 31;
    const int gp0  = blockIdx.x * POS_PER_BLOCK + wave * POS_PER_WAVE;
    if (gp0 >= total_pos) return;

    // ---- CDNA5 async global->LDS stage of x: 4 positions * 64B = 256B per wave,
    //      one b64 per lane. Wave waits on its own ASYNCcnt; no barrier needed
    //      since each wave only reads the x it staged itself.
    {
        uint32_t lds_off = (uint32_t)(uintptr_t)(&xs[(wave * POS_PER_WAVE) * 16])
                         + (uint32_t)(lane * 8);
        uint64_t gaddr   = (uint64_t)(uintptr_t)(x + (size_t)gp0 * 16)
                         + (uint64_t)(lane * 8);
        asm volatile("global_load_async_to_lds_b64 %0, %1, off"
                     :: "v"(lds_off), "v"(gaddr)
                     : "memory");
        asm volatile("s_wait_asynccnt 0x0" ::: "memory");
    }

    const int half = lane >> 4;   // 0: supplies A k-slots 0,1 (yx 0,1); 1: k-slots 2,3 (yx 2,3)
    const int mrow = lane & 15;   // A-matrix row: output index within 16-block

    for (int q = 0; q < POS_PER_WAVE; ++q) {
        const size_t gp = (size_t)(gp0 + q);

        // broadcast reads: all low lanes hit one LDS address, all high lanes another
        const v4f* xv = (const v4f*)&xs[(wave * POS_PER_WAVE + q) * 16 + half * 8];
        const v4f xk0 = xv[0];   // yx = 2*half + 0, i = 0..3
        const v4f xk1 = xv[1];   // yx = 2*half + 1, i = 0..3

        const v4f* wp = (const v4f*)(w + gp * 1024);
        float*     op = out + gp * 64;

        #pragma unroll
        for (int c = 0; c < 4; ++c) {
            const int oc = c * 16 + mrow;
            // lanes 0-15 read a contiguous 256B run (yx=0/1), lanes 16-31 likewise (yx=2/3):
            // every weight byte touched exactly once, streamed non-temporal.
            v4f wv0 = __builtin_nontemporal_load(&wp[(half * 2 + 0) * 64 + oc]);
            v4f wv1 = __builtin_nontemporal_load(&wp[(half * 2 + 1) * 64 + oc]);

            // elementwise products; the sums run on the matrix pipe below
            v4f q0 = wv0 * xk0;
            v4f q1 = wv1 * xk1;

            v8f acc = {0.f, 0.f, 0.f, 0.f, 0.f, 0.f, 0.f, 0.f};
            v2f ones; ones[0] = 1.0f; ones[1] = 1.0f;
            v2f a;
            // D[m,n] += sum_k A[m,k] * 1 : four chained K=4 WMMAs sum all 16
            // products of row m (8 from lane m, 8 from lane m+16). Exact f32.
            a[0] = q0.x; a[1] = q0.y;
            acc = __builtin_amdgcn_wmma_f32_16x16x4_f32(false, a, false, ones, (short)0, acc, false, false);
            a[0] = q0.z; a[1] = q0.w;
            acc = __builtin_amdgcn_wmma_f32_16x16x4_f32(false, a, false, ones, (short)0, acc, false, false);
            a[0] = q1.x; a[1] = q1.y;
            acc = __builtin_amdgcn_wmma_f32_16x16x4_f32(false, a, false, ones, (short)0, acc, false, false);
            a[0] = q1.z; a[1] = q1.w;
            acc = __builtin_amdgcn_wmma_f32_16x16x4_f32(false, a, false, ones, (short)0, acc, false, false);

            // D row r lives in acc[r]: lane 0 holds rows 0..7, lane 16 rows 8..15
            if (mrow == 0) {
                v4f s0; s0.x = acc[0]; s0.y = acc[1]; s0.z = acc[2]; s0.w = acc[3];
                v4f s1; s1.x = acc[4]; s1.y = acc[5]; s1.z = acc[6]; s1.w = acc[7];
                v4f* dst = (v4f*)(op + c * 16 + half * 8);
                dst[0] = s0;
                dst[1] = s1;
            }
        }
    }
}

extern "C" void kernel_launch(void* const* d_in, const int* in_sizes, int n_in,
                              void* d_out, int out_size, void* d_ws, size_t ws_size,
                              hipStream_t stream) {
    const float* x   = (const float*)d_in[0];   // [2,8,128,128,4] f32 -> flat [GP,16]
    const float* w   = (const float*)d_in[1];   // [2,8,4096,2,2,8,8,4] f32 -> flat [GP,1024]
    float*       out = (float*)d_out;           // flat [GP,64]

    const int GP = in_sizes[0] / 16;            // 65536 positions
    dim3 grid(GP / POS_PER_BLOCK);              // 2048 blocks
    dim3 block(256);                            // 8 wave32s
    predict_vectors_wmma_kernel<<<grid, block, 0, stream>>>(x, w, out, GP);
}